// DynamicQueryAllocation_1958505087203
// MI455X (gfx1250) — compile-verified
//
#include <hip/hip_runtime.h>
#include <hip/hip_bf16.h>
#include <cmath>

#define BDIM 2
#define CDIM 6
#define NDIM 4000

typedef __attribute__((ext_vector_type(2))) float v2f;
typedef __attribute__((ext_vector_type(8))) float v8f;
typedef __attribute__((ext_vector_type(4))) float f4;
typedef __attribute__((ext_vector_type(4))) unsigned char uc4;

#if __has_builtin(__builtin_amdgcn_wmma_f32_16x16x4_f32)
#define HAVE_WMMA_F32X4 1
#else
#define HAVE_WMMA_F32X4 0
#endif

// ---------------------------------------------------------------------------
// Kernel 1: geometry. One wave (32 lanes) per 16-anchor tile.
// Projection of 9 points x 6 cameras done with V_WMMA_F32_16X16X4_F32:
//   A (16x4)  = 16 anchors' homogeneous point k   (lanes 0-15: x,y ; 16-31: z,1)
//   B (4x16)  = 16 columns = (cam, out-row) pairs of the 4x4 projection mats
//   D (16x16) = projected coords; staged through LDS to regroup per (anchor,cam)
// Tile is widened to 36 columns so BOTH wmma result tiles store unconditionally
// (lanes with col>=8 of the second tile land in scratch columns 24..31 that are
// never read) -- avoids exec-mask save/restore branches around ds_store.
// ---------------------------------------------------------------------------
__global__ __launch_bounds__(32)
void geom_kernel(const float* __restrict__ anchor, const float* __restrict__ proj,
                 const int* __restrict__ image_wh,
                 float* __restrict__ sc, float* __restrict__ cd,
                 unsigned char* __restrict__ maskinfo,
                 float* __restrict__ out_tmask /* (B,N,C) as float */)
{
    __shared__ float tile[16][36];
    const int lane = threadIdx.x;
    const int blk  = blockIdx.x;
    const int b    = blk / (NDIM / 16);
    const int t    = blk % (NDIM / 16);
    const int arow = lane & 15;
    const int n    = t * 16 + arow;

    const float img_w = (float)image_wh[0];
    const float img_h = (float)image_wh[1];

    // -------- per-lane anchor -> 9 points (8 corners + center) --------
    const float* ap = anchor + ((size_t)b * NDIM + n) * 11;
    float cx = ap[0], cy = ap[1], cz = ap[2];
    float sw = fminf(expf(ap[3]), 35.f);
    float sl = fminf(expf(ap[4]), 35.f);
    float sh = fminf(expf(ap[5]), 10.f);
    float sy = ap[6], cw = ap[7];

    float px[9], py[9], pz[9];
#pragma unroll
    for (int k = 0; k < 8; ++k) {
        float d0 = (float)((k >> 2) & 1) - 0.5f;
        float d1 = (float)((k >> 1) & 1) - 0.5f;
        float d2 = (float)(k & 1) - 0.5f;
        float lx = sw * d0, ly = sl * d1, lz = sh * d2;
        px[k] = cw * lx - sy * ly + cx;
        py[k] = sy * lx + cw * ly + cy;
        pz[k] = lz + cz;
    }
    px[8] = cx; py[8] = cy; pz[8] = cz;

#if HAVE_WMMA_F32X4
    // -------- B operands: projection matrix rows, resident across k loop ----
    // column N = 4*cam + row ; lane L holds B[K][N=L&15] with K = (L<16?0:2)+vgpr
    const int q  = lane & 15;
    const int c0 = q >> 2, irow = q & 3;
    const int j0 = (lane < 16) ? 0 : 2;
    const float* pA = proj + (((size_t)b * CDIM + c0) * 4 + irow) * 4 + j0;
    v2f b1; b1.x = pA[0]; b1.y = pA[1];
    v2f b2;
    if (c0 < 2) {
        const float* pB = proj + (((size_t)b * CDIM + (4 + c0)) * 4 + irow) * 4 + j0;
        b2.x = pB[0]; b2.y = pB[1];
    } else { b2.x = 0.f; b2.y = 0.f; }
    v8f czero = {0.f, 0.f, 0.f, 0.f, 0.f, 0.f, 0.f, 0.f};
#endif

    // -------- per-(anchor,cam) accumulators; pairs p = lane + 32*s ----------
    float xmin[3], xmax[3], ymin[3], ymax[3];
    float ccu[3], ccv[3], cdep[3];
    bool  cval[3], kval[3];
#pragma unroll
    for (int s = 0; s < 3; ++s) {
        xmin[s] = 1e30f; xmax[s] = -1e30f; ymin[s] = 1e30f; ymax[s] = -1e30f;
        cval[s] = false; kval[s] = false; ccu[s] = 0.f; ccv[s] = 0.f; cdep[s] = 0.f;
    }

    for (int k = 0; k < 9; ++k) {
        __syncthreads();   // protect LDS tile from previous-iteration readers
#if HAVE_WMMA_F32X4
        v2f a;
        a.x = (lane < 16) ? px[k] : pz[k];
        a.y = (lane < 16) ? py[k] : 1.0f;
        v8f d1 = __builtin_amdgcn_wmma_f32_16x16x4_f32(false, a, false, b1,
                                                       (short)0, czero, false, false);
        v8f d2 = __builtin_amdgcn_wmma_f32_16x16x4_f32(false, a, false, b2,
                                                       (short)0, czero, false, false);
        {
            const int row0 = (lane < 16) ? 0 : 8;
            const int col  = lane & 15;
#pragma unroll
            for (int r = 0; r < 8; ++r) {
                tile[row0 + r][col] = d1[r];
                tile[row0 + r][16 + col] = d2[r];   // cols 24..31 = scratch
            }
        }
#else
        if (lane < 16) {
            tile[arow][0] = px[k]; tile[arow][1] = py[k]; tile[arow][2] = pz[k];
        }
#endif
        __syncthreads();

#pragma unroll
        for (int s = 0; s < 3; ++s) {
            const int p = lane + 32 * s;          // 0..95
            const int a_i = p / 6, c_i = p % 6;
            float X, Y, Z;
#if HAVE_WMMA_F32X4
            X = tile[a_i][c_i * 4 + 0];
            Y = tile[a_i][c_i * 4 + 1];
            Z = tile[a_i][c_i * 4 + 2];
#else
            {
                const float* P = proj + (((size_t)b * CDIM + c_i) * 4) * 4;
                float qx = tile[a_i][0], qy = tile[a_i][1], qz = tile[a_i][2];
                X = P[0] * qx + P[1] * qy + P[2]  * qz + P[3];
                Y = P[4] * qx + P[5] * qy + P[6]  * qz + P[7];
                Z = P[8] * qx + P[9] * qy + P[10] * qz + P[11];
            }
#endif
            float zc = fmaxf(Z, 1e-5f);
            float u = X / zc, v = Y / zc;
            if (k < 8) {
                bool iv = (u > 0.f) & (u < img_w) & (v > 0.f) & (v < img_h);
                kval[s] = kval[s] | ((Z > 0.f) & iv);
                xmin[s] = fminf(xmin[s], u); xmax[s] = fmaxf(xmax[s], u);
                ymin[s] = fminf(ymin[s], v); ymax[s] = fmaxf(ymax[s], v);
            } else {
                ccu[s] = u; ccv[s] = v; cdep[s] = Z;
                cval[s] = (u > 0.f) & (u < img_w) & (v > 0.f) & (v < img_h);
            }
        }
    }

    // -------- emit per (anchor,cam) results --------
#pragma unroll
    for (int s = 0; s < 3; ++s) {
        const int p = lane + 32 * s;
        const int a_i = p / 6, c_i = p % 6;
        const int nn = t * 16 + a_i;
        if (nn >= NDIM) continue;
        float x0 = fminf(fmaxf(xmin[s], 0.f), img_w);
        float x1 = fminf(fmaxf(xmax[s], 0.f), img_w);
        float y0 = fminf(fmaxf(ymin[s], 0.f), img_h);
        float y1 = fminf(fmaxf(ymax[s], 0.f), img_h);
        float su = cval[s] ? ccu[s] : 0.5f * (x0 + x1);
        float sv = cval[s] ? ccv[s] : 0.5f * (y0 + y1);
        bool  tm = cval[s] | kval[s];
        size_t idx = ((size_t)b * CDIM + c_i) * NDIM + nn;
        sc[idx * 2 + 0] = su;
        sc[idx * 2 + 1] = sv;
        cd[idx] = fabsf(cdep[s]);
        maskinfo[idx] = tm ? (cval[s] ? (unsigned char)2 : (unsigned char)1)
                           : (unsigned char)0;
        out_tmask[((size_t)b * NDIM + nn) * CDIM + c_i] = tm ? 1.f : 0.f;
    }
}

// ---------------------------------------------------------------------------
// Kernel 2: stable per-(b,c) prefix scan of mask over N (wave32 ballot scan)
// ---------------------------------------------------------------------------
__global__ __launch_bounds__(32)
void scan_kernel(const unsigned char* __restrict__ maskinfo,
                 int* __restrict__ rank, int* __restrict__ tshape)
{
    const int bc = blockIdx.x;              // b*C + c
    const int lane = threadIdx.x;
    const unsigned char* mi = maskinfo + (size_t)bc * NDIM;
    int* rk = rank + (size_t)bc * NDIM;
    int running = 0;
    for (int base = 0; base < NDIM; base += 32) {
        int n = base + lane;
        bool m = (n < NDIM) && (mi[n] != 0);
        unsigned long long bal = __ballot(m);
        int r = running + __popcll(bal & ((1ull << lane) - 1ull));
        if (m) rk[n] = r;
        running += __popcll(bal);
    }
    if (lane == 0) tshape[bc] = running;
}

// ---------------------------------------------------------------------------
// Kernel 3: meta (starts = cumsum of max-over-batch counts), small outputs,
// zero-init of scatter targets, scode/row bookkeeping init.
// ---------------------------------------------------------------------------
__global__ __launch_bounds__(256)
void meta_kernel(const int* __restrict__ tshape, int* __restrict__ starts,
                 const int M,
                 float* __restrict__ out_tshape, float* __restrict__ out_qg,
                 float* __restrict__ o_pts, float* __restrict__ o_depth,
                 int* __restrict__ row_n, unsigned char* __restrict__ rv,
                 unsigned char* __restrict__ scode)
{
    __shared__ int s_starts[8];
    __shared__ int s_tsh[BDIM * CDIM];
    if (threadIdx.x == 0) {
        int acc = 0;
        s_starts[0] = 0;
        for (int c = 0; c < CDIM; ++c) {
            int mx = 0;
            for (int b = 0; b < BDIM; ++b) mx = max(mx, tshape[b * CDIM + c]);
            acc += mx;
            s_starts[c + 1] = acc;
        }
        for (int i = 0; i < 7; ++i) starts[i] = s_starts[i];
    }
    if (threadIdx.x < BDIM * CDIM) s_tsh[threadIdx.x] = tshape[threadIdx.x];
    __syncthreads();
    if (threadIdx.x < CDIM) {
        out_qg[threadIdx.x * 2 + 0] = (float)s_starts[threadIdx.x];
        out_qg[threadIdx.x * 2 + 1] = (float)s_starts[threadIdx.x + 1];
    }
    if (threadIdx.x < BDIM * CDIM) out_tshape[threadIdx.x] = (float)s_tsh[threadIdx.x];
    for (int pos = threadIdx.x; pos < M; pos += blockDim.x) {
        int c = 0;
        while (c < CDIM - 1 && pos >= s_starts[c + 1]) ++c;   // cam_id(pos)
        for (int b = 0; b < BDIM; ++b) {
            int rowi = b * M + pos;
            bool sel = (pos - s_starts[c]) < s_tsh[b * CDIM + c];
            scode[rowi] = sel ? (unsigned char)(c + 1) : (unsigned char)0;
            row_n[rowi] = -1;
            rv[rowi] = 0;
            o_pts[(size_t)rowi * 2 + 0] = 0.f;
            o_pts[(size_t)rowi * 2 + 1] = 0.f;
            o_depth[rowi] = 0.f;
        }
    }
}

// ---------------------------------------------------------------------------
// Kernel 4: scatter compacted queries
// ---------------------------------------------------------------------------
__global__ __launch_bounds__(256)
void scatter_kernel(const float* __restrict__ sc, const float* __restrict__ cd,
                    const unsigned char* __restrict__ maskinfo,
                    const int* __restrict__ rank, const int* __restrict__ starts,
                    const int* __restrict__ image_wh, const int M,
                    float* __restrict__ o_pts, float* __restrict__ o_depth,
                    int* __restrict__ row_n, unsigned char* __restrict__ rv)
{
    int tid = blockIdx.x * blockDim.x + threadIdx.x;
    if (tid >= BDIM * CDIM * NDIM) return;
    int n  = tid % NDIM;
    int bc = tid / NDIM;
    int c = bc % CDIM, b = bc / CDIM;
    unsigned char mi = maskinfo[tid];
    if (!mi) return;
    int pos  = starts[c] + rank[tid];
    int rowi = b * M + pos;
    float iw = (float)image_wh[0], ih = (float)image_wh[1];
    o_pts[(size_t)rowi * 2 + 0] = sc[(size_t)tid * 2 + 0] / iw;
    o_pts[(size_t)rowi * 2 + 1] = sc[(size_t)tid * 2 + 1] / ih;
    o_depth[rowi] = cd[tid];
    row_n[rowi] = n;
    rv[rowi] = mi;
}

// ---------------------------------------------------------------------------
// Kernel 5: stream-fill trans_matrix / center_matrix rows (NT b128 stores)
// ---------------------------------------------------------------------------
__global__ __launch_bounds__(256)
void fill_tc_kernel(const int* __restrict__ row_n, const unsigned char* __restrict__ rv,
                    float* __restrict__ o_trans, float* __restrict__ o_center)
{
    const int blk = blockIdx.x;                 // b*M + m
    const int rn  = row_n[blk];
    const unsigned char vv = rv[blk];
    const float tv = (vv >= 1) ? 1.f : 0.f;
    const float cv = (vv == 2) ? 1.f : 0.f;
    float* tp = o_trans  + (size_t)blk * NDIM;
    float* cp = o_center + (size_t)blk * NDIM;
    const bool v4ok = ((((size_t)tp) | ((size_t)cp)) & 15) == 0;
    if (v4ok) {
        f4* tp4 = (f4*)tp; f4* cp4 = (f4*)cp;
        const int qq = rn >> 2, rr = rn & 3;
        for (int j = threadIdx.x; j < NDIM / 4; j += 256) {
            f4 tvec = {0.f, 0.f, 0.f, 0.f};
            f4 cvec = {0.f, 0.f, 0.f, 0.f};
            if (rn >= 0 && j == qq) { tvec[rr] = tv; cvec[rr] = cv; }
            __builtin_nontemporal_store(tvec, tp4 + j);
            __builtin_nontemporal_store(cvec, cp4 + j);
        }
    } else {
        for (int j = threadIdx.x; j < NDIM; j += 256) {
            __builtin_nontemporal_store((j == rn) ? tv : 0.f, tp + j);
            __builtin_nontemporal_store((j == rn) ? cv : 0.f, cp + j);
        }
    }
}

// ---------------------------------------------------------------------------
// Kernel 6: attention mask (B,M,M): 0 iff both selected & same camera group
// ---------------------------------------------------------------------------
__global__ __launch_bounds__(256)
void attn_kernel(const unsigned char* __restrict__ scode, const int M,
                 float* __restrict__ attn)
{
    const int blk = blockIdx.x;                 // b*M + i
    const int b = blk / M;
    const unsigned char si = scode[blk];
    const unsigned char* sj = scode + (size_t)b * M;
    float* row = attn + (size_t)blk * M;
    const float NEG = __int_as_float(0xff800000);   // -inf
    const bool v4ok = ((M & 3) == 0) && ((((size_t)row) | ((size_t)sj)) & 15) == 0;
    if (v4ok) {
        const uc4* sj4 = (const uc4*)sj;
        f4* row4 = (f4*)row;
        for (int j = threadIdx.x; j < (M >> 2); j += 256) {
            uc4 s4 = sj4[j];
            f4 o;
            o[0] = (si && s4[0] == si) ? 0.f : NEG;
            o[1] = (si && s4[1] == si) ? 0.f : NEG;
            o[2] = (si && s4[2] == si) ? 0.f : NEG;
            o[3] = (si && s4[3] == si) ? 0.f : NEG;
            __builtin_nontemporal_store(o, row4 + j);
        }
    } else {
        for (int j = threadIdx.x; j < M; j += 256) {
            __builtin_nontemporal_store((si && sj[j] == si) ? 0.f : NEG, row + j);
        }
    }
}

// ---------------------------------------------------------------------------
extern "C" void kernel_launch(void* const* d_in, const int* in_sizes, int n_in,
                              void* d_out, int out_size, void* d_ws, size_t ws_size,
                              hipStream_t stream)
{
    const float* anchor  = (const float*)d_in[0];
    const float* proj    = (const float*)d_in[1];
    const int*   imagewh = (const int*)d_in[2];
    float* out = (float*)d_out;

    // out_size = 2M^2 + 16006M + 48024  (B=2, N=4000, C=6) -> solve for M
    double rhs  = (double)out_size - 48024.0;
    double disc = 16006.0 * 16006.0 + 8.0 * rhs;
    long long Ml = (long long)llround((sqrt(disc) - 16006.0) / 4.0);
    if (Ml < 0) Ml = 0;
    const int M = (int)Ml;

    // output regions (flat, reference return order, all float32)
    float* o_pts    = out;                                   // B*M*2
    float* o_depth  = o_pts    + (size_t)BDIM * M * 2;       // B*M
    float* o_tmask  = o_depth  + (size_t)BDIM * M;           // B*N*C
    float* o_tshape = o_tmask  + (size_t)BDIM * NDIM * CDIM; // B*C
    float* o_trans  = o_tshape + (size_t)BDIM * CDIM;        // B*M*N
    float* o_center = o_trans  + (size_t)BDIM * M * NDIM;    // B*M*N
    float* o_qg     = o_center + (size_t)BDIM * M * NDIM;    // C*2
    float* o_attn   = o_qg     + (size_t)CDIM * 2;           // B*M*M

    // workspace carve (256B aligned chunks)
    char* w = (char*)d_ws;
    auto carve = [&](size_t bytes) {
        char* p = w;
        w += (bytes + 255) & ~(size_t)255;
        return p;
    };
    const size_t BM = (size_t)BDIM * (M > 0 ? M : 1);
    float*         sc       = (float*)carve(sizeof(float) * (size_t)BDIM * CDIM * NDIM * 2);
    float*         cdws     = (float*)carve(sizeof(float) * (size_t)BDIM * CDIM * NDIM);
    int*           rank     = (int*)carve(sizeof(int) * (size_t)BDIM * CDIM * NDIM);
    unsigned char* maskinfo = (unsigned char*)carve((size_t)BDIM * CDIM * NDIM);
    int*           tshape   = (int*)carve(sizeof(int) * BDIM * CDIM);
    int*           starts   = (int*)carve(sizeof(int) * 8);
    int*           row_n    = (int*)carve(sizeof(int) * BM);
    unsigned char* rv       = (unsigned char*)carve(BM);
    unsigned char* scode    = (unsigned char*)carve(BM);

    geom_kernel<<<BDIM * (NDIM / 16), 32, 0, stream>>>(
        anchor, proj, imagewh, sc, cdws, maskinfo, o_tmask);

    scan_kernel<<<BDIM * CDIM, 32, 0, stream>>>(maskinfo, rank, tshape);

    meta_kernel<<<1, 256, 0, stream>>>(
        tshape, starts, M, o_tshape, o_qg, o_pts, o_depth, row_n, rv, scode);

    const int total = BDIM * CDIM * NDIM;
    scatter_kernel<<<(total + 255) / 256, 256, 0, stream>>>(
        sc, cdws, maskinfo, rank, starts, imagewh, M, o_pts, o_depth, row_n, rv);

    if (M > 0) {
        fill_tc_kernel<<<BDIM * M, 256, 0, stream>>>(row_n, rv, o_trans, o_center);
        attn_kernel<<<BDIM * M, 256, 0, stream>>>(scode, M, o_attn);
    }
}